// FPSPooling_Module_50577534878087
// MI455X (gfx1250) — compile-verified
//
#include <hip/hip_runtime.h>
#include <cfloat>
#include <cstdint>

// Problem constants (from reference): N=16384 points, 3 coords + 128 feats,
// M = N/4 = 4096 FPS samples, K = 16 neighbors, out = (4096, 131) f32.
#define NPTS 16384
#define FEAT 128
#define ROW  131
#define MQ   4096
#define KNN  16
#define CH   512   // points per streamed chunk in knn kernel
#define QB   64    // queries per block in knn kernel

typedef float v2f __attribute__((ext_vector_type(2)));
typedef float v8f __attribute__((ext_vector_type(8)));

// Low 32 bits of a flat shared-aperture address are the LDS byte offset
// (CDNA5 ISA).
__device__ __forceinline__ uint32_t lds_off(const void* p) {
  return (uint32_t)(uintptr_t)p;
}

// CDNA5 async global->LDS copy (ASYNCcnt path, CDNA5 ISA).
// VDST VGPR carries the per-lane LDS byte address, VADDR the 64-bit global addr.
__device__ __forceinline__ void async_gl2lds_b32(const float* g, const void* l) {
  asm volatile("global_load_async_to_lds_b32 %0, %1, off"
               :: "v"(lds_off(l)), "v"(g)
               : "memory");
}
__device__ __forceinline__ void wait_async0() {
  asm volatile("s_wait_asynccnt 0" ::: "memory");
}

// ---------------------------------------------------------------------------
// Kernel 1: furthest point sampling. Single workgroup (serial dependence),
// coords resident in 192KB LDS (CDNA5 WGP has 320KB), min_d2 in registers.
// Writes q_points (cols 0..2) of every output row.
// ---------------------------------------------------------------------------
__global__ void __launch_bounds__(1024)
fps_kernel(const float* __restrict__ x, float* __restrict__ out) {
  extern __shared__ float sm[];
  float* sx   = sm;               // NPTS
  float* sy   = sm + NPTS;        // NPTS
  float* sz   = sm + 2 * NPTS;    // NPTS
  float* bc   = sm + 3 * NPTS;    // 4: broadcast coords of last pick
  float* redd = bc + 4;           // 32: per-wave best dist
  int*   redi = (int*)(redd + 32);// 32: per-wave best idx

  const int t = threadIdx.x;
  const int lane = t & 31, wave = t >> 5;
  const int p0 = t * 16;

  // Stage all coords into LDS with async global->LDS copies (SoA layout).
  for (int i = 0; i < 16; ++i) {
    const float* g = x + (size_t)(p0 + i) * ROW;
    async_gl2lds_b32(g + 0, sx + p0 + i);
    async_gl2lds_b32(g + 1, sy + p0 + i);
    async_gl2lds_b32(g + 2, sz + p0 + i);
  }
  wait_async0();
  __syncthreads();

  float cx[16], cy[16], cz[16], md[16];
#pragma unroll
  for (int i = 0; i < 16; ++i) {
    cx[i] = sx[p0 + i]; cy[i] = sy[p0 + i]; cz[i] = sz[p0 + i];
    md[i] = FLT_MAX;
  }
  if (t == 0) {
    bc[0] = sx[0]; bc[1] = sy[0]; bc[2] = sz[0];
    out[0] = sx[0]; out[1] = sy[0]; out[2] = sz[0];  // fps_idx[0] = 0
  }
  __syncthreads();

  for (int it = 1; it < MQ; ++it) {
    const float lx = bc[0], ly = bc[1], lz = bc[2];
    float best = -1.0f; int bidx = 0;
#pragma unroll
    for (int i = 0; i < 16; ++i) {
      // Match reference FP order exactly: (dx*dx + dy*dy) + dz*dz, no FMA.
      float dx = cx[i] - lx, dy = cy[i] - ly, dz = cz[i] - lz;
      float d2 = __fadd_rn(__fadd_rn(__fmul_rn(dx, dx), __fmul_rn(dy, dy)),
                           __fmul_rn(dz, dz));
      float m = fminf(md[i], d2);
      md[i] = m;
      if (m > best) { best = m; bidx = p0 + i; }  // ascending i => lowest idx wins ties
    }
    // wave32 argmax reduction, ties -> lowest global index (jnp.argmax semantics)
#pragma unroll
    for (int off = 16; off > 0; off >>= 1) {
      float ob = __shfl_xor(best, off, 32);
      int   oi = __shfl_xor(bidx, off, 32);
      if (ob > best || (ob == best && oi < bidx)) { best = ob; bidx = oi; }
    }
    if (lane == 0) { redd[wave] = best; redi[wave] = bidx; }
    __syncthreads();
    if (wave == 0) {
      float bb = redd[lane]; int ii = redi[lane];
#pragma unroll
      for (int off = 16; off > 0; off >>= 1) {
        float ob = __shfl_xor(bb, off, 32);
        int   oi = __shfl_xor(ii, off, 32);
        if (ob > bb || (ob == bb && oi < ii)) { bb = ob; ii = oi; }
      }
      if (lane == 0) {
        float wx = sx[ii], wy = sy[ii], wz = sz[ii];
        bc[0] = wx; bc[1] = wy; bc[2] = wz;
        float* o = out + (size_t)it * ROW;
        o[0] = wx; o[1] = wy; o[2] = wz;
      }
    }
    __syncthreads();
  }
}

// ---------------------------------------------------------------------------
// Kernel 2: KNN (top-16 by d2) via WMMA distance tiles + feature max-pool.
// Block = 256 threads (8 waves) handles 64 queries; streams N in 512-pt
// chunks (async->LDS), computes d2 = |q|^2 - 2 q.p + |p|^2 with
// V_WMMA_F32_16X16X4_F32 (A = -2*Q 16x4, B = P 4x16, C = |q|^2+|p|^2),
// stores tiles to LDS, 4-way split register top-16 selection, merge,
// then gathers neighbor features and max-pools into out cols 3..130.
// Reads query coords from out rows written by fps_kernel.
//
// Assumed WMMA register layouts (CDNA5 ISA §7.12.2):
//  A 16x4:  lane<16 -> row M=lane, v0=K0(x), v1=K1(y);
//           lane>=16 -> row M=lane-16, v0=K2(z), v1=K3(0)
//  B 4x16:  lane<16 -> col N=lane, v0=K0(x), v1=K1(y);
//           lane>=16 -> col N=lane-16, v0=K2(z), v1=K3(0)
//  C/D:     vgpr r, lane<16 -> (M=r, N=lane); lane>=16 -> (M=r+8, N=lane-16)
// ---------------------------------------------------------------------------
__global__ void __launch_bounds__(256)
knn_pool_kernel(const float* __restrict__ x, float* __restrict__ out) {
  extern __shared__ float sm[];
  float* px  = sm;                 // CH
  float* py  = px + CH;            // CH
  float* pz  = py + CH;            // CH
  float* pn2 = pz + CH;            // CH
  float* qn2 = pn2 + CH;           // QB
  float* d2b = qn2 + QB;           // QB*CH (reused for merge candidates)
  int*   nbr = (int*)(d2b + QB * CH);  // QB*KNN

  const int t = threadIdx.x, lane = t & 31, wave = t >> 5;
  const bool hi = lane >= 16; const int lr = lane & 15;
  const int qbase = blockIdx.x * QB;

  if (t < QB) {
    const float* q = out + (size_t)(qbase + t) * ROW;
    float a = q[0], b = q[1], c = q[2];
    qn2[t] = fmaf(a, a, fmaf(b, b, c * c));
  }
  __syncthreads();

  // Per-lane A operands (-2q) for the 4 query tiles, plus |q|^2 rows for C.
  float a0[4], a1[4], cq[4][8];
#pragma unroll
  for (int qt = 0; qt < 4; ++qt) {
    const float* q = out + (size_t)(qbase + qt * 16 + lr) * ROW;
    float qx = q[0], qy = q[1], qz = q[2];
    a0[qt] = hi ? (-2.0f * qz) : (-2.0f * qx);
    a1[qt] = hi ? 0.0f         : (-2.0f * qy);
#pragma unroll
    for (int r = 0; r < 8; ++r) cq[qt][r] = qn2[qt * 16 + (hi ? 8 : 0) + r];
  }

  // Register top-16 (replace-current-max scheme; arrays stay in VGPRs since
  // all indexing is unrolled-constant).
  float bd[KNN]; int bi[KNN];
#pragma unroll
  for (int s = 0; s < KNN; ++s) { bd[s] = FLT_MAX; bi[s] = 0; }
  float worst = FLT_MAX; int wslot = 0;

  const int selq = t & 63;   // query this thread selects for
  const int part = t >> 6;   // 4-way split of each chunk
  const int j0 = part * (CH / 4);

  for (int cb = 0; cb < NPTS; cb += CH) {
    // Stage chunk coords (SoA) via async global->LDS.
#pragma unroll
    for (int e = 0; e < 2; ++e) {
      int pp = t * 2 + e;
      const float* g = x + (size_t)(cb + pp) * ROW;
      async_gl2lds_b32(g + 0, px + pp);
      async_gl2lds_b32(g + 1, py + pp);
      async_gl2lds_b32(g + 2, pz + pp);
    }
    wait_async0();
    __syncthreads();
#pragma unroll
    for (int e = 0; e < 2; ++e) {
      int pp = t * 2 + e;
      float a = px[pp], b = py[pp], c = pz[pp];
      pn2[pp] = fmaf(a, a, fmaf(b, b, c * c));
    }
    __syncthreads();

    // WMMA phase: wave w computes point-tiles w*4..w*4+3 against all 4 qtiles.
#pragma unroll
    for (int i = 0; i < 4; ++i) {
      const int pt = wave * 4 + i;
      const int col = pt * 16 + lr;
      v2f B; B.x = hi ? pz[col] : px[col];
             B.y = hi ? 0.0f    : py[col];
      const float pn = pn2[col];
#pragma unroll
      for (int qt = 0; qt < 4; ++qt) {
        v2f A; A.x = a0[qt]; A.y = a1[qt];
        v8f C;
#pragma unroll
        for (int r = 0; r < 8; ++r) C[r] = cq[qt][r] + pn;
        v8f D = __builtin_amdgcn_wmma_f32_16x16x4_f32(
            false, A, false, B, (short)0, C, false, false);
        const int rowb = qt * 16 + (hi ? 8 : 0);
#pragma unroll
        for (int r = 0; r < 8; ++r) d2b[(rowb + r) * CH + col] = D[r];
      }
    }
    __syncthreads();

    // Selection: each thread scans its quarter of its query's d2 row.
    const float* drow = d2b + selq * CH;
    for (int j = j0; j < j0 + (CH / 4); ++j) {
      float d = drow[j];
      if (d < worst) {
        int gi = cb + j;
#pragma unroll
        for (int s = 0; s < KNN; ++s) if (s == wslot) { bd[s] = d; bi[s] = gi; }
        worst = bd[0]; wslot = 0;
#pragma unroll
        for (int s = 1; s < KNN; ++s) if (bd[s] > worst) { worst = bd[s]; wslot = s; }
      }
    }
    __syncthreads();
  }

  // Merge the 4 partial top-16 lists per query (candidates in reused d2b).
  float* cd = d2b;                      // QB*64 floats
  int*   ci = (int*)(d2b + QB * 64);    // QB*64 ints
#pragma unroll
  for (int s = 0; s < KNN; ++s) {
    cd[selq * 64 + part * 16 + s] = bd[s];
    ci[selq * 64 + part * 16 + s] = bi[s];
  }
  __syncthreads();
  if (t < QB) {
    float* crow = cd + t * 64; int* irow = ci + t * 64;
    for (int k = 0; k < KNN; ++k) {
      float m = crow[0]; int ms = 0;
      for (int c2 = 1; c2 < 64; ++c2) { float v = crow[c2]; if (v < m) { m = v; ms = c2; } }
      nbr[t * KNN + k] = irow[ms];
      crow[ms] = FLT_MAX;
    }
  }
  __syncthreads();

  // Feature gather + max-pool: 4 threads per query, feature f = sub + 4*j so
  // each load instruction touches 16B-contiguous runs per query row.
  const int pq = t >> 2, sub = t & 3;
  float acc[32];
#pragma unroll
  for (int j = 0; j < 32; ++j) acc[j] = -FLT_MAX;
  for (int k = 0; k < KNN; ++k) {
    const float* f = x + (size_t)nbr[pq * KNN + k] * ROW + 3;
#pragma unroll
    for (int j = 0; j < 32; ++j) acc[j] = fmaxf(acc[j], f[sub + 4 * j]);
  }
  float* o = out + (size_t)(qbase + pq) * ROW + 3;
#pragma unroll
  for (int j = 0; j < 32; ++j) o[sub + 4 * j] = acc[j];
}

// ---------------------------------------------------------------------------
extern "C" void kernel_launch(void* const* d_in, const int* in_sizes, int n_in,
                              void* d_out, int out_size, void* d_ws, size_t ws_size,
                              hipStream_t stream) {
  (void)in_sizes; (void)n_in; (void)d_ws; (void)ws_size; (void)out_size;
  const float* x = (const float*)d_in[0];
  float* out = (float*)d_out;

  // FPS: single workgroup, 192KB+ dynamic LDS (CDNA5 WGP supports up to 320KB).
  size_t fps_smem = (size_t)(3 * NPTS + 4 + 32) * sizeof(float) + 32 * sizeof(int);
  hipLaunchKernelGGL(fps_kernel, dim3(1), dim3(1024), fps_smem, stream, x, out);

  // KNN + pool: 64 blocks x 256 threads, ~140KB dynamic LDS per block.
  size_t knn_smem = (size_t)(4 * CH + QB + (size_t)QB * CH) * sizeof(float)
                  + (size_t)QB * KNN * sizeof(int);
  hipLaunchKernelGGL(knn_pool_kernel, dim3(MQ / QB), dim3(256), knn_smem, stream, x, out);
}